// RINEchoModel_75892072121104
// MI455X (gfx1250) — compile-verified
//
#include <hip/hip_runtime.h>
#include <hip/hip_bf16.h>
#include <math.h>

// Model dims
#define BB   4
#define SS   256
#define DD   256
#define LL   2
#define NNF  512      // FFN hidden
#define HH   8
#define VV   50257
#define RR   4096
#define DHH  64       // head dim (2*D/H)

typedef __attribute__((ext_vector_type(2))) float v2f;
typedef __attribute__((ext_vector_type(8))) float v8f;
typedef __attribute__((ext_vector_type(4))) unsigned int v4u;
typedef __attribute__((ext_vector_type(8))) int v8i;
typedef __attribute__((ext_vector_type(4))) int v4i;

#define TWO_PI_F 6.28318530717958648f
#define PHI_F    1.61803398874989485f

#if __has_builtin(__builtin_amdgcn_tensor_load_to_lds) && __has_builtin(__builtin_amdgcn_s_wait_tensorcnt)
#define HAVE_TDM 1
#else
#define HAVE_TDM 0
#endif

__device__ __forceinline__ float sigmoidf_(float x) {
    return 1.0f / (1.0f + expf(-x));
}

// Match jnp: pos = theta*(R/2pi) % R (non-negative), linear interp of sin/cos tables
__device__ __forceinline__ void lut_sincos(float theta, float* s, float* c) {
    const float k = (float)RR / TWO_PI_F;
    float pos = fmodf(theta * k, (float)RR);
    if (pos < 0.0f) pos += (float)RR;
    float i0f  = floorf(pos);
    float frac = pos - i0f;
    int i0 = ((int)i0f) & (RR - 1);
    int i1 = (i0 + 1) & (RR - 1);
    const float step = TWO_PI_F / (float)RR;
    float a0 = (float)i0 * step, a1 = (float)i1 * step;
    *s = (1.0f - frac) * sinf(a0) + frac * sinf(a1);
    *c = (1.0f - frac) * cosf(a0) + frac * cosf(a1);
}

// ---------------------------------------------------------------------------
// Kernel A: phase recurrence. 1024 independent (b,d) chains, 256 steps each.
// ---------------------------------------------------------------------------
__global__ void phase_scan_kernel(const int* __restrict__ ids,
                                  const float* __restrict__ emb,
                                  float* __restrict__ Xr, float* __restrict__ Xi,
                                  float* __restrict__ sp_tab, float* __restrict__ cp_tab,
                                  float* __restrict__ hr_out, float* __restrict__ hi_out) {
    int gid = blockIdx.x * blockDim.x + threadIdx.x;   // 0..B*D-1
    int b = gid / DD, d = gid % DD;
    float hr = 0.0f, hi = 0.0f;
    for (int t = 0; t < SS; ++t) {
        int id = ids[b * SS + t];
        float w  = emb[(size_t)id * (2 * DD) + d];
        float be = emb[(size_t)id * (2 * DD) + DD + d];
        float wl = 1.0f + fabsf(w);
        float tphi = fmodf((float)t * PHI_F, TWO_PI_F);
        float thr = hr / wl + be + tphi;
        float thi = hi / wl + be + tphi;
        float sr, cr, si, ci;
        lut_sincos(thr, &sr, &cr);
        lut_sincos(thi, &si, &ci);
        hr = cr * ci - sr * si;
        hi = cr * si + sr * ci;
        Xr[(size_t)(t * BB + b) * DD + d] = hr;
        Xi[(size_t)(t * BB + b) * DD + d] = hi;
        if (gid == 0) {
            float sp, cp;
            lut_sincos(tphi, &sp, &cp);
            sp_tab[t] = sp; cp_tab[t] = cp;
        }
    }
    hr_out[gid] = hr;
    hi_out[gid] = hi;
}

// ---------------------------------------------------------------------------
// Kernel C: sequential linear-attention state scan per (b,h). One workgroup
// per (b,h), 256 threads; state S (64x64) lives in registers (16/thread).
// ---------------------------------------------------------------------------
__global__ void state_scan_kernel(const float* __restrict__ Q,
                                  const float* __restrict__ Km,
                                  const float* __restrict__ Vm,
                                  const float* __restrict__ alpha,
                                  const float* __restrict__ sp_tab,
                                  const float* __restrict__ cp_tab,
                                  float* __restrict__ O,
                                  float* __restrict__ Sf, int l) {
    int bh = blockIdx.x;               // 0..B*H-1
    int b = bh / HH, h = bh % HH;
    int tid = threadIdx.x;             // 0..255
    int e  = tid & 63;                 // output column
    int d0 = (tid >> 6) * 16;          // 16-row slice of S
    float a   = alpha[l * HH + h];
    float oma = 1.0f - a;
    __shared__ float qs[64], ks[64], vs[64], red[256];
    float Sreg[16];
#pragma unroll
    for (int i = 0; i < 16; ++i) Sreg[i] = 0.0f;

    for (int t = 0; t < SS; ++t) {
        size_t base = (size_t)(t * BB + b) * NNF + h * DHH;
        if (tid < 64) {
            qs[tid] = Q[base + tid];
            ks[tid] = Km[base + tid];
            vs[tid] = Vm[base + tid];
        }
        __syncthreads();
        if (tid < 32) {   // rotate pairs (j, j+32)
            float cp = cp_tab[t], sp = sp_tab[t];
            float q0 = qs[tid], q1 = qs[tid + 32];
            qs[tid]      = q0 * cp - q1 * sp;
            qs[tid + 32] = q0 * sp + q1 * cp;
            float k0 = ks[tid], k1 = ks[tid + 32];
            ks[tid]      = k0 * cp - k1 * sp;
            ks[tid + 32] = k0 * sp + k1 * cp;
        }
        __syncthreads();
        float ve = vs[e];
        float part = 0.0f;
#pragma unroll
        for (int i = 0; i < 16; ++i) {
            int dd = d0 + i;
            Sreg[i] = oma * Sreg[i] + a * ks[dd] * ve;
            part += qs[dd] * Sreg[i];
        }
        red[tid] = part;
        __syncthreads();
        if (tid < 64) {
            float o = (red[tid] + red[tid + 64] + red[tid + 128] + red[tid + 192]) * 0.125f;
            O[base + tid] = o;
        }
        __syncthreads();
    }
    size_t sbase = (size_t)((l * BB + b) * HH + h) * (DHH * DHH);
#pragma unroll
    for (int i = 0; i < 16; ++i)
        Sf[sbase + (size_t)(d0 + i) * DHH + e] = Sreg[i];
}

// ---------------------------------------------------------------------------
// Xp = Xr + Xi, Xm = Xr - Xi  (so logits GEMM uses raw Wor/Woi panels)
// ---------------------------------------------------------------------------
__global__ void addsub_kernel(const float* __restrict__ Xr, const float* __restrict__ Xi,
                              float* __restrict__ Xp, float* __restrict__ Xm) {
    int i = blockIdx.x * blockDim.x + threadIdx.x;
    float r = Xr[i], im = Xi[i];
    Xp[i] = r + im;
    Xm[i] = r - im;
}

// ---------------------------------------------------------------------------
// Dual fp32 WMMA GEMM: C = A1@B1 + (NEGA2 ? -A2 : A2)@B2.
// Block = 8 waves = 128(M)x64(N) tile.  B1/B2 chunks (32x64) are staged
// into LDS as raw copies via the Tensor Data Mover (double-buffered:
// chunk c+1's DMA overlaps chunk c's 64 WMMAs; s_wait_tensorcnt(2) relies
// on per-wave in-order TENSORcnt completion), else manual loads.  LDS row
// stride 72 floats (TDM pad: 8 DWORDs per 64) keeps half-wave fragment
// reads on disjoint banks.  A fragments = float2 global loads (ISA 16x4
// layout).  EP: 0=store, 1=sigmoid-gate onto Xr/Xi, 2=SiLU, 3=residual
// add, 4=logits store with (t,b)->(b,t) row remap.
// ---------------------------------------------------------------------------
#define KC 32     // K chunk staged in LDS
#define RS 72     // LDS row stride in floats (64 + 8 pad)

#if HAVE_TDM
// Issue one TDM 2-D tile load: KC x 64 f32 tile at gp (row stride ldb elems)
// into LDS offset lds_off, 72-float LDS rows, OOB zero-fill via tensor dims.
__device__ __forceinline__ void tdm_issue(const float* gp, unsigned lds_byte_off,
                                          int n_rem, int k_rem, int ldb) {
    unsigned td0 = (unsigned)n_rem;
    unsigned td1 = (unsigned)k_rem;
    unsigned w0c = 0x20000u | (1u << 20) | (5u << 22) | (7u << 25);
    v8i g1 = { (int)w0c,
               (int)((td0 & 0xFFFFu) << 16),
               (int)((td0 >> 16) | ((td1 & 0xFFFFu) << 16)),
               (int)((td1 >> 16) | (64u << 16)),
               (int)(unsigned)KC,
               (int)(unsigned)ldb,
               0, 0 };
    v4i gz  = {0, 0, 0, 0};
    v8i gz8 = {0, 0, 0, 0, 0, 0, 0, 0};
    unsigned long long ga = (unsigned long long)(const void*)gp;
    v4u g0 = { 1u, lds_byte_off, (unsigned)ga,
               (unsigned)((ga >> 32) & 0x01FFFFFFull) | 0x80000000u };
    __builtin_amdgcn_tensor_load_to_lds(g0, g1, gz, gz, gz8, 0);
}
#endif

template <int NEGA2, int EP>
__global__ void dual_gemm_wmma(const float* __restrict__ A1, const float* __restrict__ A2, int lda,
                               const float* __restrict__ B1, const float* __restrict__ B2, int ldb,
                               float* __restrict__ C, int ldc,
                               float* __restrict__ gXr, float* __restrict__ gXi,
                               int N, int K) {
    __shared__ float ldsB1[2][KC * RS];
    __shared__ float ldsB2[2][KC * RS];

    int tid  = threadIdx.x;
    int lane = tid & 31;
    int wave = tid >> 5;
    int m0 = blockIdx.y * 128 + wave * 16;
    int n0 = blockIdx.x * 64;
    int arow = lane & 15;
    int asub = (lane >> 4) << 1;      // 0 or 2 (K sub-offset per ISA A-layout)
    int ncol = lane & 15;

    const float* a1p = A1 + (size_t)(m0 + arow) * lda + asub;
    const float* a2p = A2 + (size_t)(m0 + arow) * lda + asub;

    v8f zero = {0.f, 0.f, 0.f, 0.f, 0.f, 0.f, 0.f, 0.f};
    v8f acc[4] = {zero, zero, zero, zero};

    int nc = K / KC;
#if HAVE_TDM
    if (wave == 0) {   // prime the pipeline: chunk 0 -> buffer 0
        tdm_issue(B1 + n0, (unsigned)(unsigned long long)(const void*)&ldsB1[0][0],
                  N - n0, K, ldb);
        tdm_issue(B2 + n0, (unsigned)(unsigned long long)(const void*)&ldsB2[0][0],
                  N - n0, K, ldb);
    }
#endif

    for (int c = 0; c < nc; ++c) {
        int k0c = c * KC;
        __syncthreads();              // buffer (c+1)&1 free: compute c-1 done
#if HAVE_TDM
        if (wave == 0) {
            if (c + 1 < nc) {
                int kn = k0c + KC;
                tdm_issue(B1 + (size_t)kn * ldb + n0,
                          (unsigned)(unsigned long long)(const void*)&ldsB1[(c + 1) & 1][0],
                          N - n0, K - kn, ldb);
                tdm_issue(B2 + (size_t)kn * ldb + n0,
                          (unsigned)(unsigned long long)(const void*)&ldsB2[(c + 1) & 1][0],
                          N - n0, K - kn, ldb);
                __builtin_amdgcn_s_wait_tensorcnt(2);   // chunk c landed (in-order)
            } else {
                __builtin_amdgcn_s_wait_tensorcnt(0);
            }
        }
        __syncthreads();
        const float* p1 = &ldsB1[c & 1][0];
        const float* p2 = &ldsB2[c & 1][0];
#else
#pragma unroll
        for (int i = 0; i < (KC * 64) / 256; ++i) {
            int idx = tid + i * 256;
            int row = idx >> 6, col = idx & 63;
            int gc = n0 + col;
            float b1 = 0.f, b2 = 0.f;
            if (gc < N) {
                size_t goff = (size_t)(k0c + row) * ldb + gc;
                b1 = B1[goff];
                b2 = B2[goff];
                if (k0c + KC < K) {
                    __builtin_prefetch(B1 + goff + (size_t)KC * ldb, 0, 1);
                    __builtin_prefetch(B2 + goff + (size_t)KC * ldb, 0, 1);
                }
            }
            ldsB1[0][row * RS + col] = b1;
            ldsB2[0][row * RS + col] = b2;
        }
        __syncthreads();
        const float* p1 = &ldsB1[0][0];
        const float* p2 = &ldsB2[0][0];
#endif

#pragma unroll
        for (int kk = 0; kk < KC; kk += 4) {
            v2f a1 = *(const v2f*)(a1p);
            v2f a2 = *(const v2f*)(a2p);
            a1p += 4; a2p += 4;
            if (NEGA2) { a2.x = -a2.x; a2.y = -a2.y; }
            int kr = (kk + asub) * RS + ncol;
#pragma unroll
            for (int j = 0; j < 4; ++j) {
                v2f bp, bm;
                bp.x = p1[kr + j * 16];
                bp.y = p1[kr + RS + j * 16];
                bm.x = p2[kr + j * 16];
                bm.y = p2[kr + RS + j * 16];
                acc[j] = __builtin_amdgcn_wmma_f32_16x16x4_f32(
                    false, a1, false, bp, (short)0, acc[j], false, false);
                acc[j] = __builtin_amdgcn_wmma_f32_16x16x4_f32(
                    false, a2, false, bm, (short)0, acc[j], false, false);
            }
        }
    }

    int rbase = m0 + ((lane >> 4) << 3);      // C layout: VGPR v -> row v / v+8
#pragma unroll
    for (int j = 0; j < 4; ++j) {
        int col = n0 + j * 16 + ncol;
        if (col >= N) continue;
#pragma unroll
        for (int v = 0; v < 8; ++v) {
            int row = rbase + v;
            float val = acc[j][v];
            if (EP == 0) {
                C[(size_t)row * ldc + col] = val;
            } else if (EP == 1) {             // gate: x *= sigmoid(o)
                float g = sigmoidf_(val);
                if (col < DD) gXr[(size_t)row * DD + col] *= g;
                else          gXi[(size_t)row * DD + (col - DD)] *= g;
            } else if (EP == 2) {             // SiLU
                C[(size_t)row * ldc + col] = val * sigmoidf_(val);
            } else if (EP == 3) {             // residual accumulate
                C[(size_t)row * ldc + col] += val;
            } else {                          // logits: row (t*B+b) -> (b*S+t)
                int tt = row >> 2, bo = row & 3;
                C[(size_t)(bo * SS + tt) * ldc + col] = val;
            }
        }
    }
}

// ---------------------------------------------------------------------------
// Workspace layout (floats)
// ---------------------------------------------------------------------------
#define XR_OFF 0
#define XI_OFF 262144
#define XP_OFF 524288
#define XM_OFF 786432
#define Q_OFF  1048576
#define K_OFF  1572864
#define V_OFF  2097152
#define O_OFF  2621440
#define FR_OFF 3145728
#define FI_OFF 3670016
#define SP_OFF 4194304
#define CP_OFF 4194560

extern "C" void kernel_launch(void* const* d_in, const int* in_sizes, int n_in,
                              void* d_out, int out_size, void* d_ws, size_t ws_size,
                              hipStream_t stream) {
    const int*   ids   = (const int*)  d_in[0];
    const float* emb   = (const float*)d_in[1];
    const float* Wq    = (const float*)d_in[2];
    const float* Wk    = (const float*)d_in[3];
    const float* Wv    = (const float*)d_in[4];
    const float* Wo    = (const float*)d_in[5];
    const float* alpha = (const float*)d_in[6];
    const float* W1r   = (const float*)d_in[7];
    const float* W1i   = (const float*)d_in[8];
    const float* W2r   = (const float*)d_in[9];
    const float* W2i   = (const float*)d_in[10];
    const float* Wor   = (const float*)d_in[11];
    const float* Woi   = (const float*)d_in[12];

    float* ws = (float*)d_ws;
    float* Xr = ws + XR_OFF;
    float* Xi = ws + XI_OFF;
    float* Xp = ws + XP_OFF;
    float* Xm = ws + XM_OFF;
    float* Qb = ws + Q_OFF;
    float* Kb = ws + K_OFF;
    float* Vb = ws + V_OFF;
    float* Ob = ws + O_OFF;
    float* Fr = ws + FR_OFF;
    float* Fi = ws + FI_OFF;
    float* sp = ws + SP_OFF;
    float* cp = ws + CP_OFF;

    float* out    = (float*)d_out;
    float* hr_out = out + (size_t)BB * SS * VV;
    float* hi_out = hr_out + BB * DD;
    float* Sf     = hi_out + BB * DD;

    phase_scan_kernel<<<BB * DD / 256, 256, 0, stream>>>(ids, emb, Xr, Xi, sp, cp,
                                                         hr_out, hi_out);

    dim3 blk(256);
    dim3 g512(8, 8);   // N=512, M=1024
    dim3 g256(4, 8);   // N=256
    for (int l = 0; l < LL; ++l) {
        const float* Wql  = Wq  + (size_t)l * 512 * 512;
        const float* Wkl  = Wk  + (size_t)l * 512 * 512;
        const float* Wvl  = Wv  + (size_t)l * 512 * 512;
        const float* Wol  = Wo  + (size_t)l * 512 * 512;
        const float* W1rl = W1r + (size_t)l * 256 * 512;
        const float* W1il = W1i + (size_t)l * 256 * 512;
        const float* W2rl = W2r + (size_t)l * 512 * 256;
        const float* W2il = W2i + (size_t)l * 512 * 256;

        // Q/K/V = [Xr|Xi] @ W  (split W rows at 256)
        dual_gemm_wmma<0, 0><<<g512, blk, 0, stream>>>(Xr, Xi, DD,
            Wql, Wql + 256 * 512, 512, Qb, 512, nullptr, nullptr, 512, 256);
        dual_gemm_wmma<0, 0><<<g512, blk, 0, stream>>>(Xr, Xi, DD,
            Wkl, Wkl + 256 * 512, 512, Kb, 512, nullptr, nullptr, 512, 256);
        dual_gemm_wmma<0, 0><<<g512, blk, 0, stream>>>(Xr, Xi, DD,
            Wvl, Wvl + 256 * 512, 512, Vb, 512, nullptr, nullptr, 512, 256);

        // Sequential state scan (rotation + S update + o)
        state_scan_kernel<<<BB * HH, 256, 0, stream>>>(Qb, Kb, Vb, alpha, sp, cp,
                                                       Ob, Sf, l);

        // O @ Wo, fused sigmoid gate onto Xr/Xi
        dual_gemm_wmma<0, 1><<<g512, blk, 0, stream>>>(Ob, Ob + 256, 512,
            Wol, Wol + 256 * 512, 512, nullptr, 0, Xr, Xi, 512, 256);

        // Complex FFN stage 1 (+SiLU):  Fr = silu(Xr@W1r - Xi@W1i), Fi = silu(Xr@W1i + Xi@W1r)
        dual_gemm_wmma<1, 2><<<g512, blk, 0, stream>>>(Xr, Xi, DD,
            W1rl, W1il, 512, Fr, 512, nullptr, nullptr, 512, 256);
        dual_gemm_wmma<0, 2><<<g512, blk, 0, stream>>>(Xr, Xi, DD,
            W1il, W1rl, 512, Fi, 512, nullptr, nullptr, 512, 256);

        // Complex FFN stage 2 (+residual): Xr += Fr@W2r - Fi@W2i, Xi += Fr@W2i + Fi@W2r
        dual_gemm_wmma<1, 3><<<g256, blk, 0, stream>>>(Fr, Fi, 512,
            W2rl, W2il, 256, Xr, 256, nullptr, nullptr, 256, 512);
        dual_gemm_wmma<0, 3><<<g256, blk, 0, stream>>>(Fr, Fi, 512,
            W2il, W2rl, 256, Xi, 256, nullptr, nullptr, 256, 512);
    }

    // logits = (Xr+Xi)@Wor + (Xr-Xi)@Woi, rows remapped (t,b)->(b,t)
    addsub_kernel<<<(BB * SS * DD) / 256, 256, 0, stream>>>(Xr, Xi, Xp, Xm);
    dual_gemm_wmma<0, 4><<<dim3((VV + 63) / 64, 8), blk, 0, stream>>>(Xp, Xm, DD,
        Wor, Woi, VV, out, VV, nullptr, nullptr, VV, 256);
}